// StratifiedRaysampler_18270790877575
// MI455X (gfx1250) — compile-verified
//
#include <hip/hip_runtime.h>

// ---------------------------------------------------------------------------
// StratifiedRaysampler for MI455X (gfx1250, wave32).
//
//   points[ray, pt, c] = (z[pt] / d_z[ray]) * d[ray, c]     (128 pts, 3 comps)
//   lengths[ray, pt]   = z[pt]
//
// 512 MiB of output -> pure store-bandwidth bound (~23 us @ 23.3 TB/s).
// Per-ray outer product mapped onto V_WMMA_F32_16X16X4_F32:
//   A (16x4): column K=0 = 16 z values (one pt-tile), other K cols = 0
//   B (4x16): row    K=0 = w[ray, c] = d[c]/d_z for 16 rays, other rows = 0
//   C = 0  ->  D[m][n] = z[m] * w[n]  exactly (single product, zero padding)
// D is staged through LDS (padded row stride) so the global stores stay
// perfectly coalesced b128 with non-temporal hint (output >> 192MB L2).
// ---------------------------------------------------------------------------

typedef __attribute__((ext_vector_type(2))) float v2f;
typedef __attribute__((ext_vector_type(4))) float v4f;
typedef __attribute__((ext_vector_type(8))) float v8f;

#define N_RAYS   262144
#define N_PTS    128
#define MIN_D    0.1f
#define MAX_D    6.0f
#define RAYS_WG  16                 // rays handled per workgroup (one WMMA N-tile)
#define LDS_STRIDE 388              // 384 floats/ray padded: 388%64 != 0, 388*4 % 16 == 0

__device__ __forceinline__ float zval(int j) {
    const float step = (MAX_D - MIN_D) / (float)(N_PTS - 1);
    return MIN_D + (float)j * step;
}

__global__ void __launch_bounds__(128)
ray_sampler_kernel(const float* __restrict__ dirs,
                   float* __restrict__ out_pts,
                   float* __restrict__ out_len) {
    __shared__ __align__(16) float lds[RAYS_WG * LDS_STRIDE];   // ~24.3 KB

    const int tid  = threadIdx.x;        // 0..127
    const int wave = tid >> 5;           // 0..3
    const int lane = tid & 31;
    const int ray0 = blockIdx.x * RAYS_WG;

    // ---- Build B (4x16, K=0 row live): w[c] = d[c]/d_z for rays ray0..+15 --
    const int  col  = lane & 15;
    const int  ridx = ray0 + col;
    const float d0 = dirs[ridx * 3 + 0];
    const float d1 = dirs[ridx * 3 + 1];
    const float d2 = dirs[ridx * 3 + 2];
    const bool blo = (lane < 16);        // lanes 0-15 carry K=0; K=1..3 rows = 0
    v2f bx = { blo ? (d0 / d2) : 0.0f, 0.0f };
    v2f by = { blo ? (d1 / d2) : 0.0f, 0.0f };
    v2f bz = { blo ? (d2 / d2) : 0.0f, 0.0f };

    // ---- Each wave: 2 pt-tiles x 3 components = 6 WMMAs -> LDS ------------
    const int ptoff = (lane >= 16) ? 8 : 0;
#pragma unroll
    for (int t = 0; t < 2; ++t) {
        const int ptbase = (wave * 2 + t) * 16;
        // A (16x4): lanes 0-15 hold K=0 column = z-tile; K=1..3 zero.
        v2f a = { blo ? zval(ptbase + lane) : 0.0f, 0.0f };
        v8f cz = {};
#if __has_builtin(__builtin_amdgcn_wmma_f32_16x16x4_f32)
        v8f c0 = __builtin_amdgcn_wmma_f32_16x16x4_f32(false, a, false, bx,
                                                       (short)0, cz, false, false);
        v8f c1 = __builtin_amdgcn_wmma_f32_16x16x4_f32(false, a, false, by,
                                                       (short)0, cz, false, false);
        v8f c2 = __builtin_amdgcn_wmma_f32_16x16x4_f32(false, a, false, bz,
                                                       (short)0, cz, false, false);
#else
        // Fallback (should not trigger on gfx1250): scalar outer product.
        v8f c0 = cz, c1 = cz, c2 = cz;
#pragma unroll
        for (int r = 0; r < 8; ++r) {
            float zm = zval(ptbase + r + ptoff);
            c0[r] = zm * __shfl(bx.x, col, 32);
            c1[r] = zm * __shfl(by.x, col, 32);
            c2[r] = zm * __shfl(bz.x, col, 32);
        }
#endif
        // D layout: VGPR r, lanes 0-15 -> (M=r, N=lane); lanes 16-31 -> (M=r+8).
#pragma unroll
        for (int r = 0; r < 8; ++r) {
            const int pt   = ptbase + r + ptoff;
            const int base = col * LDS_STRIDE + pt * 3;
            lds[base + 0] = c0[r];
            lds[base + 1] = c1[r];
            lds[base + 2] = c2[r];
        }
    }

    // ---- Lengths: pure z broadcast, independent of LDS -> overlap before bar
    // 16 rays * 128 pts = 2048 floats = 512 float4; 128 threads x 4 iters.
    {
        v4f* lenv = (v4f*)(out_len + (size_t)ray0 * N_PTS);
#pragma unroll
        for (int it = 0; it < 4; ++it) {
            const int c  = it * 128 + tid;       // float4 index, 0..511
            const int j0 = (c & 31) * 4;         // pt within ray
            v4f v = { zval(j0), zval(j0 + 1), zval(j0 + 2), zval(j0 + 3) };
            __builtin_nontemporal_store(v, lenv + c);
        }
    }

    __syncthreads();

    // ---- Coalesced b128 NT stores: 16 rays * 384 floats = 1536 float4 -----
    {
        v4f* outv = (v4f*)(out_pts + (size_t)ray0 * (N_PTS * 3));
#pragma unroll
        for (int it = 0; it < 12; ++it) {
            const int c    = it * 128 + tid;     // 0..1535, contiguous in global
            const int ray  = c / 96;             // 96 float4 per ray
            const int off4 = c - ray * 96;
            v4f v = *(const v4f*)&lds[ray * LDS_STRIDE + off4 * 4];
            __builtin_nontemporal_store(v, outv + c);
        }
    }
}

extern "C" void kernel_launch(void* const* d_in, const int* in_sizes, int n_in,
                              void* d_out, int out_size, void* d_ws, size_t ws_size,
                              hipStream_t stream) {
    (void)in_sizes; (void)n_in; (void)out_size; (void)d_ws; (void)ws_size;
    // d_in[0] = origins (unused by the reference math), d_in[1] = directions.
    const float* dirs = (const float*)d_in[1];
    float* out_pts = (float*)d_out;                                  // [N_RAYS,128,3]
    float* out_len = out_pts + (size_t)N_RAYS * N_PTS * 3;           // [N_RAYS,128,1]

    dim3 grid(N_RAYS / RAYS_WG);   // 16384 workgroups
    dim3 block(128);               // 4 wave32s
    ray_sampler_kernel<<<grid, block, 0, stream>>>(dirs, out_pts, out_len);
}